// HGTConv_38809324486860
// MI455X (gfx1250) — compile-verified
//
#include <hip/hip_runtime.h>
#include <cstdint>

// ---------------------------------------------------------------------------
// HGTConv for MI455X (gfx1250, wave32). Memory-bound op => fp32 throughout,
// matrix work routed through V_WMMA_F32_16X16X4_F32 (exact f32 matrix pipe).
// Nodes are bucketed by ntype and edges by etype so every 16-row WMMA tile
// shares a single weight / relation matrix.
// ---------------------------------------------------------------------------

#define T_  4
#define R_  8
#define H_  4
#define D_  32
#define HD_ 128
#define LAM_ 0.3f
#define INV_SQRT_D 0.17677669529663687f   // 1/sqrt(32)

// meta layout (ints): cntN[0..3] cntE[4..11] noff[12..16] eoff[17..25]
//                     curN[26..29] curE[30..37]
#define META_INTS 40

typedef __attribute__((ext_vector_type(2))) float v2f;
typedef __attribute__((ext_vector_type(8))) float v8f;

__device__ __forceinline__ v8f wmma_f32_16x16x4(v2f a, v2f b, v8f c) {
  // D = A(16x4,f32) * B(4x16,f32) + C(16x16,f32)
  return __builtin_amdgcn_wmma_f32_16x16x4_f32(false, a, false, b, (short)0, c,
                                               false, false);
}

// order-preserving float<->uint encoding for atomicMax-based segment max
__device__ __forceinline__ unsigned fenc(float f) {
  unsigned u = __float_as_uint(f);
  return (u & 0x80000000u) ? ~u : (u | 0x80000000u);
}
__device__ __forceinline__ float fdec(unsigned u) {
  unsigned b = (u & 0x80000000u) ? (u ^ 0x80000000u) : ~u;
  return __uint_as_float(b);
}

__device__ __forceinline__ void atomicAddF(float* p, float v) {
  __hip_atomic_fetch_add(p, v, __ATOMIC_RELAXED, __HIP_MEMORY_SCOPE_AGENT);
}

// ------------------------- bucketing kernels -------------------------------

__global__ void hist_kernel(const int* __restrict__ ntype,
                            const int* __restrict__ etype, int N, int E,
                            int* __restrict__ meta) {
  int i = blockIdx.x * blockDim.x + threadIdx.x;
  if (i < N) atomicAdd(&meta[0 + ntype[i]], 1);
  if (i < E) atomicAdd(&meta[4 + etype[i]], 1);
}

__global__ void scan_kernel(int* __restrict__ meta) {
  if (threadIdx.x != 0 || blockIdx.x != 0) return;
  int a = 0;
  for (int t = 0; t < T_; ++t) { meta[12 + t] = a; meta[26 + t] = a; a += meta[0 + t]; }
  meta[12 + T_] = a;
  a = 0;
  for (int r = 0; r < R_; ++r) { meta[17 + r] = a; meta[30 + r] = a; a += meta[4 + r]; }
  meta[17 + R_] = a;
}

__global__ void scatter_kernel(const int* __restrict__ ntype,
                               const int* __restrict__ etype, int N, int E,
                               int* __restrict__ meta, int* __restrict__ norder,
                               int* __restrict__ eorder) {
  int i = blockIdx.x * blockDim.x + threadIdx.x;
  if (i < N) { int p = atomicAdd(&meta[26 + ntype[i]], 1); norder[p] = i; }
  if (i < E) { int p = atomicAdd(&meta[30 + etype[i]], 1); eorder[p] = i; }
}

// map a flat tile id onto (bucket, startIndex, endIndex) over `nb` buckets
__device__ __forceinline__ bool tile_map(const int* __restrict__ off, int nb,
                                         int tile, int& bk, int& s, int& eI) {
  int acc = 0;
  for (int b = 0; b < nb; ++b) {
    int lo = off[b], hi = off[b + 1];
    int ntl = (hi - lo + 15) >> 4;
    if (tile < acc + ntl) { bk = b; s = lo + (tile - acc) * 16; eI = hi; return true; }
    acc += ntl;
  }
  return false;
}

// ----------------- typed projections: k,q,v,pre = x @ W[t] -----------------
// block = 256 threads = 8 waves; block tile = 16 same-type nodes x 128 cols;
// wave w computes cols [16w,16w+16) for all four projections.

__global__ __launch_bounds__(256) void proj_kernel(
    const float* __restrict__ x, const float* __restrict__ Wk,
    const float* __restrict__ Wq, const float* __restrict__ Wv,
    const float* __restrict__ Wp, const int* __restrict__ norder,
    const int* __restrict__ meta, float* __restrict__ kb,
    float* __restrict__ qb, float* __restrict__ vb, float* __restrict__ pb) {
  int t, s, eI;
  if (!tile_map(meta + 12, T_, (int)blockIdx.x, t, s, eI)) return;

  __shared__ float As[16 * 132];  // padded stride vs 64 banks
  for (int e2 = threadIdx.x; e2 < 16 * 128; e2 += 256) {
    int row = e2 >> 7, col = e2 & 127;
    int idx = s + row;
    float v = 0.f;
    if (idx < eI) v = x[(size_t)norder[idx] * HD_ + col];
    As[row * 132 + col] = v;
  }
  __syncthreads();

  int lane = threadIdx.x & 31;
  int wv = threadIdx.x >> 5;
  int colB = wv * 16;
  int half = lane >> 4;   // A layout: lanes 0-15 hold K 4k+{0,1}; 16-31 hold 4k+{2,3}
  int fl = lane & 15;
  int dof = half * 2;

  v2f af[32];  // whole 16x128 A tile resident in VGPRs (64 regs)
#pragma unroll
  for (int kk = 0; kk < 32; ++kk)
    af[kk] = *(const v2f*)&As[fl * 132 + kk * 4 + dof];

  const size_t wo = (size_t)t * HD_ * HD_;
  const float* Ws[4] = {Wk + wo, Wq + wo, Wv + wo, Wp + wo};
  float* Ob[4] = {kb, qb, vb, pb};

  for (int pj = 0; pj < 4; ++pj) {
    const float* W = Ws[pj];
    v8f accv = {0.f, 0.f, 0.f, 0.f, 0.f, 0.f, 0.f, 0.f};
#pragma unroll
    for (int kk = 0; kk < 32; ++kk) {
      v2f bf;
      bf.x = W[(kk * 4 + dof) * HD_ + colB + fl];
      bf.y = W[(kk * 4 + dof + 1) * HD_ + colB + fl];
      accv = wmma_f32_16x16x4(af[kk], bf, accv);
    }
    float* O = Ob[pj];
#pragma unroll
    for (int i = 0; i < 8; ++i) {  // C/D: vgpr i -> row i (lanes<16) / 8+i
      int idx = s + half * 8 + i;
      if (idx < eI) O[(size_t)norder[idx] * HD_ + colB + fl] = accv[i];
    }
  }
}

// --------------- edge attention logits: a = (k_e R_att) . q_e --------------
// one wave per tile of 16 same-etype edges

__global__ __launch_bounds__(32) void edge_att_kernel(
    const float* __restrict__ kb, const float* __restrict__ qb,
    const float* __restrict__ rel_att, const float* __restrict__ rel_pri,
    const int* __restrict__ eorder, const int* __restrict__ meta,
    const int* __restrict__ srcI, const int* __restrict__ dstI,
    float* __restrict__ abuf, unsigned* __restrict__ amax) {
  int r, s, eI;
  if (!tile_map(meta + 17, R_, (int)blockIdx.x, r, s, eI)) return;

  int lane = threadIdx.x;
  int half = lane >> 4, fl = lane & 15, dof = half * 2;
  int idx = s + fl;
  int valid = (idx < eI) ? 1 : 0;
  int eid = eorder[valid ? idx : (eI - 1)];
  int sN = srcI[eid];
  int dN = dstI[eid];

  for (int h = 0; h < H_; ++h) {
    const float* B = rel_att + (size_t)(h * R_ + r) * D_ * D_;
    const float* kvec = kb + (size_t)sN * HD_ + h * D_;
    v2f af[8];
#pragma unroll
    for (int kk = 0; kk < 8; ++kk)
      af[kk] = *(const v2f*)(kvec + kk * 4 + dof);

    float arow = 0.f;  // logit owned by lanes fl<8 (rows half*8+fl)
    for (int c = 0; c < 2; ++c) {
      v8f accv = {0.f, 0.f, 0.f, 0.f, 0.f, 0.f, 0.f, 0.f};
#pragma unroll
      for (int kk = 0; kk < 8; ++kk) {
        v2f bf;
        bf.x = B[(kk * 4 + dof) * D_ + c * 16 + fl];
        bf.y = B[(kk * 4 + dof + 1) * D_ + c * 16 + fl];
        accv = wmma_f32_16x16x4(af[kk], bf, accv);
      }
#pragma unroll
      for (int i = 0; i < 8; ++i) {
        int rsel = half ? (8 + i) : i;            // row index of accv[i]
        int dN_r = __shfl(dN, rsel, 32);
        float qv = qb[(size_t)dN_r * HD_ + h * D_ + c * 16 + fl];
        float p = accv[i] * qv;
        p += __shfl_xor(p, 1, 32);
        p += __shfl_xor(p, 2, 32);
        p += __shfl_xor(p, 4, 32);
        p += __shfl_xor(p, 8, 32);                // half-wave row reduce
        if (fl == i) arow += p;
      }
    }
    arow *= rel_pri[h * R_ + r] * INV_SQRT_D;
    int srcl = half ? (fl + 8) : fl;              // lane holding our row's edge
    int eid_r = __shfl(eid, srcl, 32);
    int dN_r = __shfl(dN, srcl, 32);
    int val_r = __shfl(valid, srcl, 32);
    if (fl < 8 && val_r) {
      abuf[(size_t)eid_r * H_ + h] = arow;
      atomicMax(&amax[(size_t)dN_r * H_ + h], fenc(arow));
    }
  }
}

// ----------------- softmax numerator + denominator accumulate --------------

__global__ void norm_kernel(const int* __restrict__ dstI,
                            float* __restrict__ abuf,
                            const unsigned* __restrict__ amax,
                            float* __restrict__ denom, int E) {
  int e = blockIdx.x * blockDim.x + threadIdx.x;
  if (e >= E) return;
  int d = dstI[e];
#pragma unroll
  for (int h = 0; h < H_; ++h) {
    float am = fdec(amax[(size_t)d * H_ + h]);
    float ex = __expf(abuf[(size_t)e * H_ + h] - am);
    abuf[(size_t)e * H_ + h] = ex;
    atomicAddF(&denom[(size_t)d * H_ + h], ex);
  }
}

// -------- msg = v_e @ R_msg (recomputed, WMMA) + weighted scatter-add ------

__global__ __launch_bounds__(32) void edge_agg_kernel(
    const float* __restrict__ vb, const float* __restrict__ rel_msg,
    const int* __restrict__ eorder, const int* __restrict__ meta,
    const int* __restrict__ srcI, const int* __restrict__ dstI,
    const float* __restrict__ abuf, const float* __restrict__ denom,
    float* __restrict__ aggb) {
  int r, s, eI;
  if (!tile_map(meta + 17, R_, (int)blockIdx.x, r, s, eI)) return;

  int lane = threadIdx.x;
  int half = lane >> 4, fl = lane & 15, dof = half * 2;
  int idx = s + fl;
  int valid = (idx < eI) ? 1 : 0;
  int eid = eorder[valid ? idx : (eI - 1)];
  int sN = srcI[eid];
  int dN = dstI[eid];

  for (int h = 0; h < H_; ++h) {
    const float* B = rel_msg + (size_t)(h * R_ + r) * D_ * D_;
    const float* vvec = vb + (size_t)sN * HD_ + h * D_;
    v2f af[8];
#pragma unroll
    for (int kk = 0; kk < 8; ++kk)
      af[kk] = *(const v2f*)(vvec + kk * 4 + dof);

    for (int c = 0; c < 2; ++c) {
      v8f accv = {0.f, 0.f, 0.f, 0.f, 0.f, 0.f, 0.f, 0.f};
#pragma unroll
      for (int kk = 0; kk < 8; ++kk) {
        v2f bf;
        bf.x = B[(kk * 4 + dof) * D_ + c * 16 + fl];
        bf.y = B[(kk * 4 + dof + 1) * D_ + c * 16 + fl];
        accv = wmma_f32_16x16x4(af[kk], bf, accv);
      }
#pragma unroll
      for (int i = 0; i < 8; ++i) {
        int rsel = half ? (8 + i) : i;
        int dN_r = __shfl(dN, rsel, 32);
        int eid_r = __shfl(eid, rsel, 32);
        int val_r = __shfl(valid, rsel, 32);
        float ex = abuf[(size_t)eid_r * H_ + h];
        float den = denom[(size_t)dN_r * H_ + h];
        float wgt = ex / den;
        if (val_r)
          atomicAddF(&aggb[(size_t)dN_r * HD_ + h * D_ + c * 16 + fl],
                     accv[i] * wgt);
      }
    }
  }
}

// --------- out = ((lam*agg + (1-lam)*pre) @ W_a[t])*alpha + x*(1-alpha) ----

__global__ __launch_bounds__(256) void final_kernel(
    const float* __restrict__ x, const float* __restrict__ Wa,
    const float* __restrict__ skip, const float* __restrict__ aggb,
    const float* __restrict__ preb, const int* __restrict__ norder,
    const int* __restrict__ meta, float* __restrict__ out) {
  int t, s, eI;
  if (!tile_map(meta + 12, T_, (int)blockIdx.x, t, s, eI)) return;

  __shared__ float As[16 * 132];
  for (int e2 = threadIdx.x; e2 < 16 * 128; e2 += 256) {
    int row = e2 >> 7, col = e2 & 127;
    int idx = s + row;
    float v = 0.f;
    if (idx < eI) {
      size_t o = (size_t)norder[idx] * HD_ + col;
      v = LAM_ * aggb[o] + (1.0f - LAM_) * preb[o];
    }
    As[row * 132 + col] = v;
  }
  __syncthreads();

  int lane = threadIdx.x & 31;
  int wv = threadIdx.x >> 5;
  int colB = wv * 16;
  int half = lane >> 4;
  int fl = lane & 15;
  int dof = half * 2;

  v2f af[32];
#pragma unroll
  for (int kk = 0; kk < 32; ++kk)
    af[kk] = *(const v2f*)&As[fl * 132 + kk * 4 + dof];

  const float* W = Wa + (size_t)t * HD_ * HD_;
  v8f accv = {0.f, 0.f, 0.f, 0.f, 0.f, 0.f, 0.f, 0.f};
#pragma unroll
  for (int kk = 0; kk < 32; ++kk) {
    v2f bf;
    bf.x = W[(kk * 4 + dof) * HD_ + colB + fl];
    bf.y = W[(kk * 4 + dof + 1) * HD_ + colB + fl];
    accv = wmma_f32_16x16x4(af[kk], bf, accv);
  }

  float alpha = 1.0f / (1.0f + __expf(-skip[t]));
#pragma unroll
  for (int i = 0; i < 8; ++i) {
    int idx = s + half * 8 + i;
    if (idx < eI) {
      size_t o = (size_t)norder[idx] * HD_ + colB + fl;
      out[o] = accv[i] * alpha + x[o] * (1.0f - alpha);
    }
  }
}

// ---------------------------------------------------------------------------

extern "C" void kernel_launch(void* const* d_in, const int* in_sizes, int n_in,
                              void* d_out, int out_size, void* d_ws,
                              size_t ws_size, hipStream_t stream) {
  const float* x       = (const float*)d_in[0];
  const float* W_k     = (const float*)d_in[1];
  const float* W_q     = (const float*)d_in[2];
  const float* W_v     = (const float*)d_in[3];
  const float* W_pre   = (const float*)d_in[4];
  const float* W_a     = (const float*)d_in[5];
  const float* rel_att = (const float*)d_in[6];
  const float* rel_msg = (const float*)d_in[7];
  const float* rel_pri = (const float*)d_in[8];
  const float* skip    = (const float*)d_in[9];
  const int* ntype = (const int*)d_in[10];
  const int* etype = (const int*)d_in[11];
  const int* srcI  = (const int*)d_in[12];
  const int* dstI  = (const int*)d_in[13];

  int N = in_sizes[0] / HD_;
  int E = in_sizes[12];
  float* out = (float*)d_out;

  // workspace carve-up
  size_t fN = (size_t)N * HD_;
  float* kb   = (float*)d_ws;
  float* qb   = kb + fN;
  float* vb   = qb + fN;
  float* pb   = vb + fN;
  float* abuf = pb + fN;                       // E*H logits -> exp values
  int* norder = (int*)(abuf + (size_t)E * H_);
  int* eorder = norder + N;
  uintptr_t zp = ((uintptr_t)(eorder + E) + 255) & ~(uintptr_t)255;
  float* aggb     = (float*)zp;                // N*HD (zeroed)
  unsigned* amax  = (unsigned*)(aggb + fN);    // N*H  (zero == encoded -max)
  float* denom    = (float*)(amax + (size_t)N * H_);  // N*H (zeroed)
  int* meta       = (int*)(denom + (size_t)N * H_);   // counters (zeroed)
  size_t zbytes = (char*)(meta + META_INTS) - (char*)aggb;
  hipMemsetAsync(aggb, 0, zbytes, stream);     // graph-capture legal

  int mx = (N > E) ? N : E;
  hist_kernel<<<(mx + 255) / 256, 256, 0, stream>>>(ntype, etype, N, E, meta);
  scan_kernel<<<1, 1, 0, stream>>>(meta);
  scatter_kernel<<<(mx + 255) / 256, 256, 0, stream>>>(ntype, etype, N, E, meta,
                                                       norder, eorder);
  int ntiles = (N + 15) / 16 + T_;
  int etiles = (E + 15) / 16 + R_;
  proj_kernel<<<ntiles, 256, 0, stream>>>(x, W_k, W_q, W_v, W_pre, norder, meta,
                                          kb, qb, vb, pb);
  edge_att_kernel<<<etiles, 32, 0, stream>>>(kb, qb, rel_att, rel_pri, eorder,
                                             meta, srcI, dstI, abuf, amax);
  norm_kernel<<<(E + 255) / 256, 256, 0, stream>>>(dstI, abuf, amax, denom, E);
  edge_agg_kernel<<<etiles, 32, 0, stream>>>(vb, rel_msg, eorder, meta, srcI,
                                             dstI, abuf, denom, aggb);
  final_kernel<<<ntiles, 256, 0, stream>>>(x, W_a, skip, aggb, pb, norder, meta,
                                           out);
}